// ALGN_70300024701472
// MI455X (gfx1250) — compile-verified
//
#include <hip/hip_runtime.h>

typedef float v2f __attribute__((ext_vector_type(2)));
typedef float v4f __attribute__((ext_vector_type(4)));
typedef float v8f __attribute__((ext_vector_type(8)));

#define N_USER  100000
#define N_ITEM  50000
#define N_NODES 150000
#define N_EDGES 2000000
#define EMB     64
#define BATCH   2048

// ---------------------------------------------------------------------------
// init: cur = acc = concat(user_emb, item_emb), vectorized float4
// ---------------------------------------------------------------------------
__global__ void lgcn_init(const float* __restrict__ ue, const float* __restrict__ ie,
                          float* __restrict__ cur, float* __restrict__ acc) {
    long i4 = (long)blockIdx.x * blockDim.x + threadIdx.x;   // float4 index
    const long total4 = (long)N_NODES * EMB / 4;
    if (i4 >= total4) return;
    const long usplit4 = (long)N_USER * EMB / 4;
    v4f v = (i4 < usplit4) ? ((const v4f*)ue)[i4] : ((const v4f*)ie)[i4 - usplit4];
    ((v4f*)cur)[i4] = v;
    ((v4f*)acc)[i4] = v;
}

// ---------------------------------------------------------------------------
// zero a node-embedding buffer (float4)
// ---------------------------------------------------------------------------
__global__ void lgcn_zero(float* __restrict__ buf) {
    long i4 = (long)blockIdx.x * blockDim.x + threadIdx.x;
    if (i4 >= (long)N_NODES * EMB / 4) return;
    ((v4f*)buf)[i4] = (v4f){0.f, 0.f, 0.f, 0.f};
}

// ---------------------------------------------------------------------------
// SpMM: one wave (32 lanes) per edge, each lane owns 2 consecutive floats.
// next[rows[e]] += vals[e] * cur[cols[e]]  via native global_atomic_add_f32.
// ---------------------------------------------------------------------------
__global__ void lgcn_spmm(const float* __restrict__ vals, const int* __restrict__ rows,
                          const int* __restrict__ cols, const float* __restrict__ cur,
                          float* __restrict__ nxt) {
    unsigned tid  = blockIdx.x * blockDim.x + threadIdx.x;
    unsigned e    = tid >> 5;            // wave-uniform edge id
    if (e >= N_EDGES) return;
    unsigned lane = tid & 31u;
    float    w    = vals[e];
    long     cb   = (long)cols[e] * EMB + lane * 2;
    long     rb   = (long)rows[e] * EMB + lane * 2;
    v2f m = *(const v2f*)(cur + cb);     // global_load_b64, row-contiguous
    __hip_atomic_fetch_add(nxt + rb + 0, w * m.x, __ATOMIC_RELAXED, __HIP_MEMORY_SCOPE_AGENT);
    __hip_atomic_fetch_add(nxt + rb + 1, w * m.y, __ATOMIC_RELAXED, __HIP_MEMORY_SCOPE_AGENT);
}

// ---------------------------------------------------------------------------
// acc += src (float4)
// ---------------------------------------------------------------------------
__global__ void lgcn_accum(float* __restrict__ acc, const float* __restrict__ src) {
    long i4 = (long)blockIdx.x * blockDim.x + threadIdx.x;
    if (i4 >= (long)N_NODES * EMB / 4) return;
    v4f a = ((const v4f*)acc)[i4];
    v4f s = ((const v4f*)src)[i4];
    ((v4f*)acc)[i4] = (v4f){a.x + s.x, a.y + s.y, a.z + s.z, a.w + s.w};
}

// ---------------------------------------------------------------------------
// gather batch rows, apply the 1/(L+1)=0.25 layer-mean scale to each side
// ---------------------------------------------------------------------------
__global__ void lgcn_gather(const float* __restrict__ acc, const int* __restrict__ users,
                            const int* __restrict__ items, float* __restrict__ ug,
                            float* __restrict__ ig) {
    int t = blockIdx.x * blockDim.x + threadIdx.x;     // over BATCH*EMB
    if (t >= BATCH * EMB) return;
    int b = t >> 6, k = t & 63;
    ug[t] = acc[(long)users[b] * EMB + k] * 0.25f;
    ig[t] = acc[((long)N_USER + items[b]) * EMB + k] * 0.25f;
}

// ---------------------------------------------------------------------------
// Scoring GEMM: out = sigmoid(U @ I^T), U,I: [2048,64] f32.
// One wave per 16x16 C-tile; K=64 covered by 16x V_WMMA_F32_16X16X4_F32.
//
// Per ISA 16x16x4 f32 layouts:
//   A (16x4, MxK): lane L<16 holds A[M=L][K=0..1] in v0,v1; L>=16 holds K=2..3.
//   B (4x16, KxN): lane L<16 holds B[K=0..1][N=L];  L>=16 holds B[K=2..3][N=L-16].
// With B[k][n] = I[n][k], both operands are the SAME pattern per lane:
//   row = tile*16 + (lane&15), cols = 4*s + 2*(lane>>4) + {0,1}  -> one b64 load.
//   C/D: lane group h=lane>>4: VGPR r holds C[M = r + 8*h][N = lane&15].
// ---------------------------------------------------------------------------
__global__ void lgcn_gemm_sigmoid(const float* __restrict__ U, const float* __restrict__ I,
                                  float* __restrict__ out) {
    const int tileN = blockIdx.x;
    const int tileM = blockIdx.y;
    const int lane  = threadIdx.x;        // 0..31, full wave, EXEC all-ones
    const int r     = lane & 15;
    const int h     = lane >> 4;          // 0 or 1

    const float* uRow = U + (long)(tileM * 16 + r) * EMB + 2 * h;
    const float* iRow = I + (long)(tileN * 16 + r) * EMB + 2 * h;

    v8f c = {};
#pragma unroll
    for (int s = 0; s < 16; ++s) {
        v2f a = *(const v2f*)(uRow + 4 * s);
        v2f b = *(const v2f*)(iRow + 4 * s);
        // (neg_a, A, neg_b, B, c_mod, C, reuse_a, reuse_b)
        c = __builtin_amdgcn_wmma_f32_16x16x4_f32(false, a, false, b,
                                                  (short)0, c, false, false);
    }

    const int n = tileN * 16 + r;
#pragma unroll
    for (int rv = 0; rv < 8; ++rv) {
        const int m = tileM * 16 + rv + 8 * h;
        float x = c[rv];
        out[(long)m * BATCH + n] = 1.0f / (1.0f + __expf(-x));
    }
}

// ---------------------------------------------------------------------------
// launch
// ---------------------------------------------------------------------------
extern "C" void kernel_launch(void* const* d_in, const int* in_sizes, int n_in,
                              void* d_out, int out_size, void* d_ws, size_t ws_size,
                              hipStream_t stream) {
    const float* user_emb = (const float*)d_in[0];
    const float* item_emb = (const float*)d_in[1];
    const float* adj_vals = (const float*)d_in[2];
    const int*   adj_rows = (const int*)d_in[3];
    const int*   adj_cols = (const int*)d_in[4];
    const int*   users    = (const int*)d_in[5];
    const int*   items    = (const int*)d_in[6];
    float*       out      = (float*)d_out;

    const size_t nodeElems = (size_t)N_NODES * EMB;
    float* buf0 = (float*)d_ws;
    float* buf1 = buf0 + nodeElems;
    float* acc  = buf1 + nodeElems;
    float* ug   = acc + nodeElems;
    float* ig   = ug + (size_t)BATCH * EMB;

    const int  T      = 256;
    const long vec4   = (long)nodeElems / 4;
    const int  gVec   = (int)((vec4 + T - 1) / T);
    const long eth    = (long)N_EDGES * 32;
    const int  gEdge  = (int)((eth + T - 1) / T);

    lgcn_init<<<gVec, T, 0, stream>>>(user_emb, item_emb, buf0, acc);

    float* cur = buf0;
    float* nxt = buf1;
    for (int l = 0; l < 3; ++l) {
        lgcn_zero <<<gVec, T, 0, stream>>>(nxt);
        lgcn_spmm <<<gEdge, T, 0, stream>>>(adj_vals, adj_rows, adj_cols, cur, nxt);
        lgcn_accum<<<gVec, T, 0, stream>>>(acc, nxt);
        float* tmp = cur; cur = nxt; nxt = tmp;
    }

    lgcn_gather<<<(BATCH * EMB + T - 1) / T, T, 0, stream>>>(acc, users, items, ug, ig);

    dim3 gGemm(BATCH / 16, BATCH / 16);   // 128 x 128 tiles
    lgcn_gemm_sigmoid<<<gGemm, 32, 0, stream>>>(ug, ig, out);
}